// DPSV_Loss_2843268350538
// MI455X (gfx1250) — compile-verified
//
#include <hip/hip_runtime.h>
#include <hip/hip_bf16.h>

#define NUM_CLASSES 16
#define B_SZ 32
#define G_SZ 50
#define N_TOT 21504          // 128*128 + 64*64 + 32*32
#define TOTAL (B_SZ * N_TOT) // 688128
#define BLK 256
#define NBLK (TOTAL / BLK)   // 2688 (fallback: 1 anchor/thread)
#define NBLK4 (TOTAL / (BLK * 4)) // 672 (fast path: 4 anchors/thread)
#define MATCH_NONE 0x7FFFFFFF

typedef __attribute__((ext_vector_type(2))) float v2f;
typedef __attribute__((ext_vector_type(8))) float v8f;

#if defined(__has_builtin)
#if __has_builtin(__builtin_amdgcn_wmma_f32_16x16x4_f32)
#define USE_WMMA_REDUCE 1
#endif
#endif

// ---------------------------------------------------------------------------
// Wave32 cross-lane f32 sum. WMMA path: A = ones(16x4), B holds per-lane value
// in VGPR0 (VGPR1 = 0)  =>  D[m][n] = sum_k B[k][n] = acc[n] + acc[n+16],
// replicated over rows, so every lane's d[0] = acc[lane&15] + acc[(lane&15)+16].
// Four xor-shuffles finish the 16 -> 1 sum (full f32, fixed order).
// ---------------------------------------------------------------------------
__device__ __forceinline__ float wave_sum32(float v) {
#ifdef USE_WMMA_REDUCE
    v2f a; a.x = 1.0f; a.y = 1.0f;
    v2f b; b.x = v;    b.y = 0.0f;
    v8f c = {};
    v8f d = __builtin_amdgcn_wmma_f32_16x16x4_f32(
        /*neg_a=*/false, a, /*neg_b=*/false, b,
        /*c_mod=*/(short)0, c, /*reuse_a=*/false, /*reuse_b=*/false);
    float s = d[0];
    s += __shfl_xor(s, 1, 32);
    s += __shfl_xor(s, 2, 32);
    s += __shfl_xor(s, 4, 32);
    s += __shfl_xor(s, 8, 32);
    return s;
#else
    float s = v;
    s += __shfl_xor(s, 16, 32);
    s += __shfl_xor(s, 8, 32);
    s += __shfl_xor(s, 4, 32);
    s += __shfl_xor(s, 2, 32);
    s += __shfl_xor(s, 1, 32);
    return s;
#endif
}

// ---------------------------------------------------------------------------
// Init match array to "no match" (b128 stores).
// ---------------------------------------------------------------------------
__global__ void dpsv_init_match(int4* __restrict__ match4) {
    int i = blockIdx.x * BLK + threadIdx.x;
    if (i < TOTAL / 4) match4[i] = make_int4(MATCH_NONE, MATCH_NONE, MATCH_NONE, MATCH_NONE);
}

// ---------------------------------------------------------------------------
// Scatter pass: each (b,g) label marks the (at most ~1 per level) cells whose
// center lies strictly inside (gt - r, gt + r), r = stride/2. Candidate cell
// is floor(gt/stride); a +/-1 window is re-verified with the *exact* f32
// expressions from the reference so the predicate is bit-identical.
// atomicMin implements "first matching label" (argmax-of-first-true).
// ---------------------------------------------------------------------------
__global__ void dpsv_scatter_match(const float* __restrict__ labels,
                                   int* __restrict__ match) {
    int i = blockIdx.x * blockDim.x + threadIdx.x;
    if (i >= B_SZ * G_SZ) return;
    int b = i / G_SZ;
    int g = i % G_SZ;
    const float* lab = labels + (size_t)i * 5;
    float X = lab[0], Y = lab[1];

    const int   Ws[3] = {128, 64, 32};
    const float ss[3] = {8.0f, 16.0f, 32.0f};
    int off = 0;
    for (int l = 0; l < 3; ++l) {
        float s = ss[l];
        float r = 0.5f * s;
        int   W = Ws[l];
        int ix0 = (int)floorf(X / s);
        int iy0 = (int)floorf(Y / s);
        for (int dy = -1; dy <= 1; ++dy) {
            int iy = iy0 + dy;
            if (iy < 0 || iy >= W) continue;
            float cy  = ((float)iy + 0.5f) * s;
            float b_t = cy - (Y - r);
            float b_b = Y + r - cy;
            for (int dx = -1; dx <= 1; ++dx) {
                int ix = ix0 + dx;
                if (ix < 0 || ix >= W) continue;
                float cx  = ((float)ix + 0.5f) * s;
                float b_l = cx - (X - r);
                float b_r = X + r - cx;
                float mn = fminf(fminf(b_l, b_t), fminf(b_r, b_b));
                if (mn > 0.0f)
                    atomicMin(&match[(size_t)b * N_TOT + off + iy * W + ix], g);
            }
        }
        off += W * W;
    }
}

// ---------------------------------------------------------------------------
// Foreground-anchor loss terms (rare path; obj term handled by caller).
// base points at channel 0 of this anchor; channel c lives at base[c*HW].
// ---------------------------------------------------------------------------
__device__ __forceinline__ void fg_loss(
    const float* __restrict__ base, int HW, int W, float s, int hw, int b, int m,
    const float* __restrict__ labels, float acc[6]) {
    const float* lab = labels + ((size_t)b * G_SZ + m) * 5;
    __builtin_prefetch(lab, 0, 3);   // global_prefetch_b8 (CDNA5 path)

    float gx = (float)(hw % W);
    float gy = (float)(hw / W);
    float v0 = base[0];
    float v1 = base[(size_t)1 * HW];
    float v2 = base[(size_t)2 * HW];
    float v3 = base[(size_t)3 * HW];
    float px = (v0 + gx) * s;
    float py = (v1 + gy) * s;
    acc[0] += fabsf((lab[0] - px) * (1.0f / 1024.0f));
    acc[1] += fabsf((lab[1] - py) * (1.0f / 1024.0f));
    acc[2] += fabsf(lab[2] - v2) * 10.0f;
    acc[3] += fabsf(lab[3] - v3) * 10.0f;

    int cidx = (int)lab[4];
    float cls = 0.0f;
    #pragma unroll
    for (int c = 0; c < NUM_CLASSES; ++c) {
        float p   = base[(size_t)(5 + c) * HW];
        float lp  = fmaxf(__logf(p), -100.0f);
        float l1p = fmaxf(__logf(1.0f - p), -100.0f);
        cls += (c == cidx) ? -lp : -l1p;
    }
    acc[5] += cls;
}

// ---------------------------------------------------------------------------
// Deterministic block reduction of 6 partials (fixed-order shfl tree + LDS).
// ---------------------------------------------------------------------------
__device__ __forceinline__ void block_reduce_store(float acc[6],
                                                   float* __restrict__ partials) {
    #pragma unroll
    for (int c = 0; c < 6; ++c) {
        float x = acc[c];
        x += __shfl_xor(x, 16, 32);
        x += __shfl_xor(x, 8, 32);
        x += __shfl_xor(x, 4, 32);
        x += __shfl_xor(x, 2, 32);
        x += __shfl_xor(x, 1, 32);
        acc[c] = x;
    }
    __shared__ float wsum[BLK / 32][6];
    int wave = threadIdx.x >> 5;
    int lane = threadIdx.x & 31;
    if (lane == 0) {
        #pragma unroll
        for (int c = 0; c < 6; ++c) wsum[wave][c] = acc[c];
    }
    __syncthreads();
    if (threadIdx.x == 0) {
        #pragma unroll
        for (int c = 0; c < 6; ++c) {
            float ssum = 0.0f;
            for (int w = 0; w < BLK / 32; ++w) ssum += wsum[w][c];
            partials[(size_t)blockIdx.x * 6 + c] = ssum;
        }
    }
}

__device__ __forceinline__ void locate_anchor(
    int n, const float* in0, const float* in1, const float* in2,
    const float*& in, int& HW, int& W, float& s, int& hw) {
    if (n < 16384)      { in = in0; HW = 16384; W = 128; s = 8.0f;  hw = n; }
    else if (n < 20480) { in = in1; HW = 4096;  W = 64;  s = 16.0f; hw = n - 16384; }
    else                { in = in2; HW = 1024;  W = 32;  s = 32.0f; hw = n - 20480; }
}

// ---------------------------------------------------------------------------
// Main pass (fast path): 4 consecutive anchors per thread. A 256-thread block
// covers 1024 aligned anchors => uniform (batch, level) per block; obj plane
// and match array are streamed as b128 loads (~512 B per wave instruction).
// Background anchors touch only the obj plane (~2.75 MB instead of 57.8 MB).
// ---------------------------------------------------------------------------
__global__ void dpsv_loss_main_v4(const float* __restrict__ in0,
                                  const float* __restrict__ in1,
                                  const float* __restrict__ in2,
                                  const float* __restrict__ labels,
                                  const int* __restrict__ match,
                                  float* __restrict__ partials) {
    int j  = blockIdx.x * BLK + threadIdx.x;   // < TOTAL/4 by construction
    int i4 = j * 4;
    int b  = i4 / N_TOT;
    int n  = i4 % N_TOT;
    const float* in; int HW, W, hw; float s;
    locate_anchor(n, in0, in1, in2, in, HW, W, s, hw);
    const float* base = in + ((size_t)b * (5 + NUM_CLASSES)) * HW + hw;

    const float4 obj4 = *(const float4*)(base + (size_t)4 * HW);
    const int4   m4   = *(const int4*)(match + i4);
    float objv[4] = {obj4.x, obj4.y, obj4.z, obj4.w};
    int   mm[4]   = {m4.x, m4.y, m4.z, m4.w};

    float acc[6] = {0, 0, 0, 0, 0, 0};
    #pragma unroll
    for (int k = 0; k < 4; ++k) {
        if (mm[k] == MATCH_NONE) {
            acc[4] += __logf(1.0f - objv[k] + 1e-8f);
        } else {
            acc[4] += 1000.0f * __logf(objv[k] + 1e-8f);
            fg_loss(base + k, HW, W, s, hw + k, b, mm[k], labels, acc);
        }
    }
    block_reduce_store(acc, partials);
}

// ---------------------------------------------------------------------------
// Fallback (tiny workspace): per-anchor 50-label first-match loop.
// ---------------------------------------------------------------------------
__global__ void dpsv_loss_fallback(const float* __restrict__ in0,
                                   const float* __restrict__ in1,
                                   const float* __restrict__ in2,
                                   const float* __restrict__ labels,
                                   float* __restrict__ partials) {
    int i = blockIdx.x * BLK + threadIdx.x;
    int b = i / N_TOT;
    int n = i % N_TOT;
    const float* in; int HW, W, hw; float s;
    locate_anchor(n, in0, in1, in2, in, HW, W, s, hw);
    const float* base = in + ((size_t)b * (5 + NUM_CLASSES)) * HW + hw;

    float gx = (float)(hw % W), gy = (float)(hw / W);
    float cx = (gx + 0.5f) * s, cy = (gy + 0.5f) * s, r = 0.5f * s;
    const float* labB = labels + (size_t)b * G_SZ * 5;
    int m = MATCH_NONE;
    for (int g = 0; g < G_SZ; ++g) {
        float X = labB[g * 5 + 0], Y = labB[g * 5 + 1];
        float b_l = cx - (X - r), b_r = X + r - cx;
        float b_t = cy - (Y - r), b_b = Y + r - cy;
        if (fminf(fminf(b_l, b_t), fminf(b_r, b_b)) > 0.0f) { m = g; break; }
    }
    float obj = base[(size_t)4 * HW];
    float acc[6] = {0, 0, 0, 0, 0, 0};
    if (m == MATCH_NONE) {
        acc[4] += __logf(1.0f - obj + 1e-8f);
    } else {
        acc[4] += 1000.0f * __logf(obj + 1e-8f);
        fg_loss(base, HW, W, s, hw, b, m, labels, acc);
    }
    block_reduce_store(acc, partials);
}

// ---------------------------------------------------------------------------
// Final reduction: one wave32; per-lane strided sums (fixed order) then a
// WMMA-based cross-lane sum. Emits v_wmma_f32_16x16x4_f32.
// ---------------------------------------------------------------------------
__global__ void dpsv_finalize(const float* __restrict__ partials, int nblk,
                              float* __restrict__ out) {
    int lane = threadIdx.x;   // blockDim.x == 32, one wave, EXEC all ones
    float acc[6] = {0, 0, 0, 0, 0, 0};
    for (int k = lane; k < nblk; k += 32) {
        #pragma unroll
        for (int c = 0; c < 6; ++c) acc[c] += partials[(size_t)k * 6 + c];
    }
    float tot[6];
    #pragma unroll
    for (int c = 0; c < 6; ++c) tot[c] = wave_sum32(acc[c]);
    if (lane == 0) {
        float asum_total = tot[0] + tot[1] + tot[2] + tot[3];
        float loss_vec = asum_total * 10000.0f;
        float loss_obj = -tot[4];
        float loss_cls = tot[5] * 500.0f;
        out[0] = loss_obj + loss_vec + loss_cls;
        out[1] = tot[0];
        out[2] = tot[1];
        out[3] = tot[2];
        out[4] = tot[3];
    }
}

extern "C" void kernel_launch(void* const* d_in, const int* in_sizes, int n_in,
                              void* d_out, int out_size, void* d_ws, size_t ws_size,
                              hipStream_t stream) {
    const float* in0    = (const float*)d_in[0];
    const float* in1    = (const float*)d_in[1];
    const float* in2    = (const float*)d_in[2];
    const float* labels = (const float*)d_in[3];
    float* out = (float*)d_out;

    const size_t match_bytes = (size_t)TOTAL * sizeof(int);
    const size_t part_bytes  = (size_t)NBLK * 6 * sizeof(float);

    if (ws_size >= match_bytes + part_bytes) {
        int*   match    = (int*)d_ws;
        float* partials = (float*)((char*)d_ws + match_bytes);
        dpsv_init_match<<<NBLK4, BLK, 0, stream>>>((int4*)match);
        dpsv_scatter_match<<<(B_SZ * G_SZ + BLK - 1) / BLK, BLK, 0, stream>>>(labels, match);
        dpsv_loss_main_v4<<<NBLK4, BLK, 0, stream>>>(in0, in1, in2, labels, match, partials);
        dpsv_finalize<<<1, 32, 0, stream>>>(partials, NBLK4, out);
    } else {
        float* partials = (float*)d_ws;
        dpsv_loss_fallback<<<NBLK, BLK, 0, stream>>>(in0, in1, in2, labels, partials);
        dpsv_finalize<<<1, 32, 0, stream>>>(partials, NBLK, out);
    }
}